// Postprocess_13426067767451
// MI455X (gfx1250) — compile-verified
//
#include <hip/hip_runtime.h>
#include <hip/hip_bf16.h>
#include <stdint.h>

// ---------------------------------------------------------------------------
// Problem constants (from the reference)
// ---------------------------------------------------------------------------
#define NB      16          // batch
#define NCH     84          // channels (4 box + 80 class)
#define NA      33600       // anchors
#define NTOP    1024        // TOPK
#define SCORE_THR 0.005f
// IoU > 0.5  <=>  inter > 0.5*(areaI+areaJ-inter)  <=>  3*inter > areaI+areaJ

#ifndef __has_builtin
#define __has_builtin(x) 0
#endif

#define GLB_AS __attribute__((address_space(1)))
#define LDS_AS __attribute__((address_space(3)))

// CDNA5 async global->LDS copy.  The builtin takes typed int* pointers
// (addrspace-qualified); AS-qualified pointers also convert implicitly to
// generic parameters, so these casts are safe for either declaration.
#if __has_builtin(__builtin_amdgcn_global_load_async_to_lds_b32)
#define ASYNC_CP(dst, src) __builtin_amdgcn_global_load_async_to_lds_b32( \
    (GLB_AS int *)(src), (LDS_AS int *)(dst), 0, 0)
#define HAVE_ASYNC 1
#else
#define ASYNC_CP(dst, src) (*(dst) = *(src))
#define HAVE_ASYNC 0
#endif

#if __has_builtin(__builtin_amdgcn_s_wait_asynccnt)
#define AWAIT(n) __builtin_amdgcn_s_wait_asynccnt(n)
#else
#define AWAIT(n) asm volatile("s_wait_asynccnt " #n ::: "memory")
#endif

typedef float v2f __attribute__((ext_vector_type(2)));
typedef float v8f __attribute__((ext_vector_type(8)));

// ---------------------------------------------------------------------------
// K1: per-anchor max/argmax over 80 classes.  Streams 180.6 MB (the whole
// cost of this problem: ~7.8us at 23.3 TB/s).  Anchor-contiguous addresses are
// perfectly coalesced; the 80-deep channel loop is double-buffered through LDS
// with async-to-LDS copies (ASYNCcnt pipelining).
// key = bits(score) if score>thr else 0   (positive f32 bits are uint-ordered)
// aux = anchor | (class<<16)
// ---------------------------------------------------------------------------
#define K1_TILES 132   // ceil(33600/256)

__global__ __launch_bounds__(256) void k1_scores(const float *__restrict__ x,
                                                 unsigned *__restrict__ keys,
                                                 unsigned *__restrict__ aux) {
  int b = blockIdx.x / K1_TILES;
  int tile = blockIdx.x % K1_TILES;
  int tid = threadIdx.x;
  int a = tile * 256 + tid;
  int aa = (a < NA) ? a : (NA - 1);           // clamp tail (result discarded)
  const float *base = x + (size_t)b * NCH * NA + aa;

  __shared__ float buf[2][256];

  ASYNC_CP(&buf[0][tid], base + (size_t)4 * NA);   // prologue: c = 4
  float m = -3.4e38f;
  int cls = 0;
  for (int c = 4; c < NCH; ++c) {
    if (c < NCH - 1) {
      ASYNC_CP(&buf[(c + 1) & 1][tid], base + (size_t)(c + 1) * NA);
      AWAIT(1);                                // oldest copy has landed
    } else {
      AWAIT(0);
    }
    float v = buf[c & 1][tid];
    if (v > m) { m = v; cls = c - 4; }         // strict '>' => first max wins
  }
  if (a < NA) {
    keys[(size_t)b * NA + a] = (m > SCORE_THR) ? __float_as_uint(m) : 0u;
    aux[(size_t)b * NA + a]  = (unsigned)a | ((unsigned)cls << 16);
  }
}

// ---------------------------------------------------------------------------
// K2: exact 1024-th largest key per batch via MSD radix select (12/12/8 bits).
// One workgroup per batch; 16 KB LDS histogram; hierarchical suffix scan.
// ---------------------------------------------------------------------------
__global__ __launch_bounds__(1024) void k2_thresh(const unsigned *__restrict__ keys,
                                                  unsigned *__restrict__ Tout) {
  int b = blockIdx.x, tid = threadIdx.x;
  const unsigned *kb = keys + (size_t)b * NA;

  __shared__ unsigned hist[4096];
  __shared__ unsigned gsum[64];
  __shared__ unsigned sh_digit, sh_need;

  const unsigned prefMaskA[3] = {0x00000000u, 0xFFF00000u, 0xFFFFFF00u};
  const int      shiftA[3]    = {20, 8, 0};
  const int      nbA[3]       = {4096, 4096, 256};
  const int      groupsA[3]   = {64, 64, 16};

  unsigned prefix = 0, need = NTOP;
  for (int p = 0; p < 3; ++p) {
    unsigned prefMask = prefMaskA[p];
    int shift = shiftA[p], nb = nbA[p], groups = groupsA[p], gsz = nb / groups;

    for (int i = tid; i < nb; i += 1024) hist[i] = 0;
    __syncthreads();
    for (int i = tid; i < NA; i += 1024) {
      unsigned k = kb[i];
      if ((k & prefMask) == prefix)
        atomicAdd(&hist[(k >> shift) & (unsigned)(nb - 1)], 1u);
    }
    __syncthreads();
    if (tid < groups) {
      unsigned s = 0;
      for (int i = 0; i < gsz; ++i) s += hist[tid * gsz + i];
      gsum[tid] = s;
    }
    __syncthreads();
    if (tid == 0) {
      unsigned acc = 0;
      int G = 0;
      for (int g = groups - 1; g >= 0; --g) {          // descending cumulative
        if (acc + gsum[g] >= need) { G = g; break; }
        acc += gsum[g];
      }
      int D = G * gsz;
      for (int d = G * gsz + gsz - 1; d >= G * gsz; --d) {
        if (acc + hist[d] >= need) { D = d; break; }
        acc += hist[d];
      }
      sh_digit = (unsigned)D;
      sh_need = need - acc;                            // #(==digit) still wanted
    }
    __syncthreads();
    prefix |= (sh_digit << shift);
    need = sh_need;
    __syncthreads();                                   // hist reused next pass
  }
  if (tid == 0) Tout[b] = prefix;
}

// ---------------------------------------------------------------------------
// K3: ordered compaction: all key>T (index-ascending), then key==T until 1024
// slots are filled (index-ascending => matches top_k tie-breaking).
// Wave32 ballot + popcount prefix sums.
// ---------------------------------------------------------------------------
__global__ __launch_bounds__(1024) void k3_compact(const unsigned *__restrict__ keys,
                                                   const unsigned *__restrict__ aux,
                                                   const unsigned *__restrict__ Tarr,
                                                   unsigned *__restrict__ selKey,
                                                   unsigned *__restrict__ selAux) {
  int b = blockIdx.x, tid = threadIdx.x;
  int lane = tid & 31, wid = tid >> 5;
  unsigned T = Tarr[b];
  const unsigned *kb = keys + (size_t)b * NA;
  const unsigned *xb = aux + (size_t)b * NA;
  unsigned *sk = selKey + b * NTOP;
  unsigned *sa = selAux + b * NTOP;

  __shared__ unsigned warpCnt[32], warpBase[32], baseCnt;
  if (tid == 0) baseCnt = 0;
  __syncthreads();

  for (int phase = 0; phase < 2; ++phase) {
    for (int start = 0; start < NA; start += 1024) {
      int i = start + tid;
      bool p = false;
      unsigned k = 0, ax = 0;
      if (i < NA) {
        k = kb[i]; ax = xb[i];
        p = (phase == 0) ? (k > T) : (k == T);
      }
      unsigned bal = __builtin_amdgcn_ballot_w32(p);
      unsigned rank = (unsigned)__builtin_popcount(bal & ((1u << lane) - 1u));
      if (lane == 0) warpCnt[wid] = (unsigned)__builtin_popcount(bal);
      __syncthreads();
      if (tid == 0) {
        unsigned acc = baseCnt;
        for (int w = 0; w < 32; ++w) { warpBase[w] = acc; acc += warpCnt[w]; }
        baseCnt = acc;
      }
      __syncthreads();
      if (p) {
        unsigned pos = warpBase[wid] + rank;
        if (pos < NTOP) { sk[pos] = k; sa[pos] = ax; }
      }
      __syncthreads();
    }
  }
}

// ---------------------------------------------------------------------------
// K4: bitonic sort of the 1024 candidates in LDS (score desc, index asc),
// then gather box coords for the sorted order.
// ---------------------------------------------------------------------------
__global__ __launch_bounds__(1024) void k4_sort(const float *__restrict__ x,
                                                unsigned *__restrict__ selKey,
                                                unsigned *__restrict__ selAux,
                                                float *__restrict__ selBox) {
  int b = blockIdx.x, tid = threadIdx.x;
  __shared__ unsigned sk[NTOP], sa[NTOP];
  sk[tid] = selKey[b * NTOP + tid];
  sa[tid] = selAux[b * NTOP + tid];
  __syncthreads();

  for (int k = 2; k <= NTOP; k <<= 1) {
    for (int j = k >> 1; j > 0; j >>= 1) {
      int ixj = tid ^ j;
      if (ixj > tid) {
        unsigned k1 = sk[tid], k2 = sk[ixj], a1 = sa[tid], a2 = sa[ixj];
        // "x before y" in final order: higher key, tie -> lower anchor index
        bool b21 = (k2 > k1) || (k2 == k1 && (a2 & 0xFFFFu) < (a1 & 0xFFFFu));
        bool b12 = (k1 > k2) || (k1 == k2 && (a1 & 0xFFFFu) < (a2 & 0xFFFFu));
        bool doSwap = ((tid & k) == 0) ? b21 : b12;
        if (doSwap) { sk[tid] = k2; sk[ixj] = k1; sa[tid] = a2; sa[ixj] = a1; }
      }
      __syncthreads();
    }
  }

  unsigned ax = sa[tid];
  unsigned anchor = ax & 0xFFFFu;
  const float *xb = x + (size_t)b * NCH * NA;
  float *bo = selBox + ((size_t)b * NTOP + tid) * 4;
#pragma unroll
  for (int c = 0; c < 4; ++c) bo[c] = xb[(size_t)c * NA + anchor];
  selKey[b * NTOP + tid] = sk[tid];
  selAux[b * NTOP + tid] = ax;
}

// ---------------------------------------------------------------------------
// K5: NMS.  Per batch: build a 1024x1024 suppression bitmask (1024x32 u32 =
// 128 KB in the 320 KB WGP LDS).  The union term area_i+area_j of each 16x16
// tile is a rank-2 outer product computed with V_WMMA_F32_16X16X4_F32:
//   A(16x4) = [area_i, 1, 0, 0],  B(4x16) = [1; area_j; 0; 0]  => D = a_i+a_j.
// WMMA C/D layout: VGPR r -> row r (lanes 0-15) and row 8+r (lanes 16-31),
// column = lane&15 -- each lane then finishes its 8 IoU tests with VALU and
// wave32 ballots pack the mask words.  The greedy pass is the classic serial
// bitmask sweep on one wave (ds_bpermute broadcast, no barriers).
// ---------------------------------------------------------------------------
__global__ __launch_bounds__(1024) void k5_nms(const unsigned *__restrict__ selKey,
                                               const unsigned *__restrict__ selAux,
                                               const float *__restrict__ selBox,
                                               float *__restrict__ out) {
  int b = blockIdx.x, tid = threadIdx.x;
  int lane = tid & 31, wid = tid >> 5;

  extern __shared__ unsigned lds[];
  unsigned *mask = lds;                        // 1024*32 words = 128 KB
  float *bx = (float *)(lds + NTOP * 32);      // x1
  float *by = bx + NTOP;                       // y1
  float *bX = by + NTOP;                       // x2
  float *bY = bX + NTOP;                       // y2
  float *ara = bY + NTOP;                      // raw area (no clip, per ref)
  unsigned *validW = (unsigned *)(ara + NTOP); // 32 words
  unsigned *keepW = validW + 32;               // 32 words

  // ---- phase 1: stage boxes/areas, ballot validity per 32-slot word -------
  unsigned key = selKey[b * NTOP + tid];
  unsigned ax  = selAux[b * NTOP + tid];
  const float *bb = selBox + ((size_t)b * NTOP + tid) * 4;
  float x1 = bb[0], y1 = bb[1], x2 = bb[2], y2 = bb[3];
  bx[tid] = x1; by[tid] = y1; bX[tid] = x2; bY[tid] = y2;
  ara[tid] = (x2 - x1) * (y2 - y1);
  unsigned vb = __builtin_amdgcn_ballot_w32(key != 0u);
  if (lane == 0) validW[wid] = vb;             // wave wid == word wid
  __syncthreads();

  // ---- phase 2: suppression mask, 2048 (i_tile, jword) tasks over 32 waves
  for (int s = 0; s < 64; ++s) {
    int task = wid * 64 + s;
    int it = task >> 5;                        // i tile 0..63 (rows it*16..+15)
    int jw = task & 31;                        // j word 0..31 (cols jw*32..+31)

    bool lo = (lane < 16);
    v2f a;  a.x = lo ? ara[it * 16 + lane] : 0.0f; a.y = lo ? 1.0f : 0.0f;
    v2f bA; bA.x = lo ? 1.0f : 0.0f; bA.y = lo ? ara[jw * 32 + lane] : 0.0f;
    v2f bBm; bBm.x = bA.x;          bBm.y = lo ? ara[jw * 32 + 16 + lane] : 0.0f;
    v8f cz = {};
    v8f dA = __builtin_amdgcn_wmma_f32_16x16x4_f32(false, a, false, bA, (short)0,
                                                   cz, false, false);
    v8f dB = __builtin_amdgcn_wmma_f32_16x16x4_f32(false, a, false, bBm, (short)0,
                                                   cz, false, false);

    int colA = jw * 32 + (lane & 15);
    int colB = colA + 16;
    float jxA = bx[colA], jyA = by[colA], jXA = bX[colA], jYA = bY[colA];
    float jxB = bx[colB], jyB = by[colB], jXB = bX[colB], jYB = bY[colB];
    int mrow0 = it * 16 + ((lane >> 4) << 3);  // +0 (lanes<16) or +8

#pragma unroll
    for (int r = 0; r < 8; ++r) {
      int m = mrow0 + r;
      float ix = bx[m], iy = by[m], iX = bX[m], iY = bY[m];
      float wA = fmaxf(fminf(iX, jXA) - fmaxf(ix, jxA), 0.0f);
      float hA = fmaxf(fminf(iY, jYA) - fmaxf(iy, jyA), 0.0f);
      float inA = wA * hA;
      bool cA = (inA > 0.0f) && (3.0f * inA > dA[r]);   // IoU > 0.5
      float wB = fmaxf(fminf(iX, jXB) - fmaxf(ix, jxB), 0.0f);
      float hB = fmaxf(fminf(iY, jYB) - fmaxf(iy, jyB), 0.0f);
      float inB = wB * hB;
      bool cB = (inB > 0.0f) && (3.0f * inB > dB[r]);
      unsigned balA = __builtin_amdgcn_ballot_w32(cA);
      unsigned balB = __builtin_amdgcn_ballot_w32(cB);
      if (lane == 0) {
        mask[(it * 16 + r) * 32 + jw]     = (balA & 0xFFFFu) | (balB << 16);
        mask[(it * 16 + 8 + r) * 32 + jw] = (balA >> 16) | (balB & 0xFFFF0000u);
      }
    }
  }
  __syncthreads();

  // ---- phase 3: serial greedy sweep on wave 0 (lane w owns removal word w)
  if (tid < 32) {
    unsigned remv = 0u, kw = 0u;
    for (int i = 0; i < NTOP; ++i) {
      int ws = i >> 5, bp = i & 31;
      unsigned rm = (unsigned)__builtin_amdgcn_ds_bpermute(ws << 2, (int)remv);
      bool removed = (rm >> bp) & 1u;
      bool vv = (validW[ws] >> bp) & 1u;
      if (!removed && vv) {                    // uniform across the wave
        remv |= mask[i * 32 + lane];
        if (lane == ws) kw |= (1u << bp);
      }
    }
    keepW[lane] = kw;
  }
  __syncthreads();

  // ---- phase 4: outputs ---------------------------------------------------
  bool kept = (keepW[tid >> 5] >> (tid & 31)) & 1u;
  size_t ob = ((size_t)b * NTOP + tid) * 6;
  float sc = __uint_as_float(key);
  float cf = (float)((ax >> 16) & 0xFFu);
  out[ob + 0] = kept ? x1 : 0.0f;
  out[ob + 1] = kept ? y1 : 0.0f;
  out[ob + 2] = kept ? x2 : 0.0f;
  out[ob + 3] = kept ? y2 : 0.0f;
  out[ob + 4] = kept ? sc : 0.0f;
  out[ob + 5] = kept ? cf : 0.0f;
  out[(size_t)NB * NTOP * 6 + (size_t)b * NTOP + tid] = kept ? 1.0f : 0.0f;
}

// ---------------------------------------------------------------------------
extern "C" void kernel_launch(void *const *d_in, const int *in_sizes, int n_in,
                              void *d_out, int out_size, void *d_ws, size_t ws_size,
                              hipStream_t stream) {
  (void)in_sizes; (void)n_in; (void)out_size; (void)ws_size;
  const float *x = (const float *)d_in[0];
  float *out = (float *)d_out;

  // workspace carve-up (~4.7 MB)
  unsigned *keys   = (unsigned *)d_ws;                 // NB*NA
  unsigned *aux    = keys + (size_t)NB * NA;           // NB*NA
  unsigned *Tarr   = aux + (size_t)NB * NA;            // NB (padded to 64)
  unsigned *selKey = Tarr + 64;                        // NB*NTOP
  unsigned *selAux = selKey + (size_t)NB * NTOP;       // NB*NTOP
  float    *selBox = (float *)(selAux + (size_t)NB * NTOP); // NB*NTOP*4

  k1_scores<<<NB * K1_TILES, 256, 0, stream>>>(x, keys, aux);
  k2_thresh<<<NB, 1024, 0, stream>>>(keys, Tarr);
  k3_compact<<<NB, 1024, 0, stream>>>(keys, aux, Tarr, selKey, selAux);
  k4_sort<<<NB, 1024, 0, stream>>>(x, selKey, selAux, selBox);

  size_t ldsBytes = (size_t)NTOP * 32 * 4   // mask
                  + (size_t)NTOP * 5 * 4    // bx,by,bX,bY,area
                  + 64 * 4;                 // validW + keepW
  k5_nms<<<NB, 1024, ldsBytes, stream>>>(selKey, selAux, selBox, out);
}